// SoftDTWBarycenter_11501922419396
// MI455X (gfx1250) — compile-verified
//
#include <hip/hip_runtime.h>
#include <hip/hip_bf16.h>

// SoftDTW barycenter loss for MI455X (gfx1250, wave32).
// Phase 1: D = x2 + z2 - 2*X.Z^T via V_WMMA_F32_16X16X4_F32, stored in a
//          SKEWED layout Dsk[n][(i+j)%T][i] (bijective) so the DP phase reads
//          each anti-diagonal as one contiguous 2KB run. GEMM workgroups cover
//          16x512 row stripes and stage through LDS so the skewed writeout is
//          64B-per-thread contiguous (no write amplification). D (64MB) stays
//          resident in the 192MB L2.
// Phase 2: anti-diagonal soft-DTW DP, one 512-thread workgroup per batch, LDS
//          triple-buffered diagonals, D values prefetched 2 steps ahead
//          (issued before the barrier), hardware exp/log softmin.
// Phase 3: tiny deterministic weighted reduction.

typedef __attribute__((ext_vector_type(2))) float v2f;
typedef __attribute__((ext_vector_type(8))) float v8f;

#define BIGF 1e10f
constexpr int N_ = 64;     // batch
constexpr int T_ = 512;    // sequence length
constexpr int D_ = 16;     // feature dim
constexpr int SROW = 516;  // padded LDS row stride (floats) to avoid bank conflicts

__device__ __forceinline__ float softmin3(float a, float b, float c) {
  // gamma == 1:  -log(e^-a + e^-b + e^-c), min-shifted for stability.
  float m = fminf(a, fminf(b, c));
  float s = __expf(m - a) + __expf(m - b) + __expf(m - c);
  return m - __logf(s);
}

// ---------------------------------------------------------------------------
// Row norms: x2[n*T+i] = ||X[n,i,:]||^2 ; z2[j] = ||Z[j,:]||^2
// ---------------------------------------------------------------------------
__global__ __launch_bounds__(256) void norms_kernel(const float* __restrict__ X,
                                                    const float* __restrict__ Z,
                                                    float* __restrict__ x2,
                                                    float* __restrict__ z2) {
  int idx = blockIdx.x * blockDim.x + threadIdx.x;
  if (idx < N_ * T_) {
    const float* p = X + (size_t)idx * D_;
    float s = 0.f;
#pragma unroll
    for (int c = 0; c < D_; ++c) s = fmaf(p[c], p[c], s);
    x2[idx] = s;
  } else if (idx < N_ * T_ + T_) {
    int j = idx - N_ * T_;
    const float* p = Z + (size_t)j * D_;
    float s = 0.f;
#pragma unroll
    for (int c = 0; c < D_; ++c) s = fmaf(p[c], p[c], s);
    z2[j] = s;
  }
}

// ---------------------------------------------------------------------------
// WMMA GEMM over a 16x512 row stripe per workgroup (16 waves, 2 tiles/wave).
// Tile math: 4 chained 16x16x4 f32 WMMAs (K=16); epilogue D = x2 + z2 - 2*acc
// goes to LDS; then a skewed, sector-aligned writeout:
//   Dsk[n][s][i] = D[i][(s-i) mod T],  thread t==s writes 16 floats (64B).
// A(16x4 f32) operand: lane = 16*half + l -> a = {A[l][2*half], A[l][2*half+1]}
// B(4x16 f32) operand: b = {B[2*half][l],  B[2*half+1][l]}  (B[k][j] = Z[j][k])
// C/D(16x16 f32):      c[r] = D[r + 8*half][l]
// ---------------------------------------------------------------------------
__global__ __launch_bounds__(T_) void gemm_wmma_kernel(const float* __restrict__ X,
                                                       const float* __restrict__ Z,
                                                       const float* __restrict__ x2,
                                                       const float* __restrict__ z2,
                                                       float* __restrict__ Dsk) {
  __shared__ float S[16 * SROW];  // 16 rows x 512 cols, padded stride

  const int n    = blockIdx.y;
  const int i0   = blockIdx.x * 16;      // stripe rows i0..i0+15
  const int tid  = threadIdx.x;
  const int wave = tid >> 5;
  const int lane = tid & 31;
  const int half = lane >> 4;
  const int l    = lane & 15;

  const float* Xi = X + ((size_t)n * T_ + i0 + l) * D_;  // A row l of stripe

#pragma unroll
  for (int t = 0; t < 2; ++t) {
    const int j0 = (wave * 2 + t) * 16;
    const float* Zj = Z + (size_t)(j0 + l) * D_;  // B col l of tile

    v8f acc = {};
#pragma unroll
    for (int kk = 0; kk < 4; ++kk) {
      const int k0 = kk * 4 + half * 2;
      v2f a, b;
      a.x = Xi[k0];
      a.y = Xi[k0 + 1];
      b.x = Zj[k0];
      b.y = Zj[k0 + 1];
      acc = __builtin_amdgcn_wmma_f32_16x16x4_f32(
          /*neg_a=*/false, a, /*neg_b=*/false, b,
          /*c_mod=*/(short)0, acc, /*reuse_a=*/false, /*reuse_b=*/false);
    }

    const float z2v = z2[j0 + l];
#pragma unroll
    for (int r = 0; r < 8; ++r) {
      const int m = half * 8 + r;  // local row in stripe
      const float x2v = x2[(size_t)n * T_ + i0 + m];
      S[m * SROW + j0 + l] = fmaf(-2.0f, acc[r], x2v + z2v);
    }
  }
  __syncthreads();

  // Skewed writeout: thread tid handles s = tid; 16 consecutive floats (64B).
  const int s = tid;
  float vals[16];
#pragma unroll
  for (int m = 0; m < 16; ++m) {
    const int jm = (s - i0 - m) & (T_ - 1);  // column for row i0+m at skew s
    vals[m] = S[m * SROW + jm];
  }
  float4* dst = (float4*)(Dsk + (size_t)n * T_ * T_ + (size_t)s * T_ + i0);
#pragma unroll
  for (int q = 0; q < 4; ++q) {
    dst[q] = make_float4(vals[4 * q], vals[4 * q + 1], vals[4 * q + 2], vals[4 * q + 3]);
  }
}

// ---------------------------------------------------------------------------
// Soft-DTW wavefront DP over the skewed D. One workgroup per batch.
// Thread i owns row i; step k reads Dsk[n][(k-2)%T][i] -- fully coalesced and
// always in-bounds (invalid lanes read the aliased diagonal, masked off).
// D values are prefetched 2 steps ahead, issued before the barrier.
// ---------------------------------------------------------------------------
__global__ __launch_bounds__(T_) void dp_kernel(const float* __restrict__ Dsk,
                                                const float* __restrict__ w,
                                                float* __restrict__ partials) {
  __shared__ float R[3][T_];
  const int n = blockIdx.x;
  const int i = threadIdx.x;
  const float* Dn = Dsk + (size_t)n * T_ * T_;

  R[0][i] = BIGF;
  R[1][i] = BIGF;

  float dp0 = Dn[i];        // D for step k=2  (s=0)
  float dp1 = Dn[T_ + i];   // D for step k=3  (s=1)
  __syncthreads();

  float last = BIGF;
  for (int k = 2; k <= 2 * T_; ++k) {
    const int cur = k % 3, p1 = (k - 1) % 3, p2 = (k - 2) % 3;
    const int j = k - 2 - i;
    float nv = BIGF;
    if (j >= 0 && j < T_) {
      float rl = R[p1][i];                                           // R[i, j-1]
      float ru = (i == 0) ? BIGF : R[p1][i - 1];                     // R[i-1, j]
      float rd = (i == 0) ? ((k == 2) ? 0.f : BIGF) : R[p2][i - 1];  // R[i-1,j-1]
      nv = dp0 + softmin3(ru, rl, rd);
    }
    R[cur][i] = nv;
    dp0 = dp1;
    dp1 = Dn[(size_t)(k & (T_ - 1)) * T_ + i];  // prefetch for step k+2 (s = k)
    __syncthreads();
    last = nv;
  }
  if (i == T_ - 1) partials[n] = w[n] * last;  // R[T][T]
}

// ---------------------------------------------------------------------------
// Fallback (workspace too small for D): fused DP that recomputes D on the fly.
// Z staged in LDS (32KB), X row held in registers.
// ---------------------------------------------------------------------------
__global__ __launch_bounds__(T_) void dp_fused_kernel(const float* __restrict__ X,
                                                      const float* __restrict__ Z,
                                                      const float* __restrict__ w,
                                                      float* __restrict__ partials) {
  __shared__ float Zs[T_ * D_];
  __shared__ float z2s[T_];
  __shared__ float R[3][T_];
  const int n = blockIdx.x;
  const int i = threadIdx.x;

  float xr[D_];
  float x2i = 0.f;
  const float* xp = X + ((size_t)n * T_ + i) * D_;
#pragma unroll
  for (int c = 0; c < D_; ++c) {
    xr[c] = xp[c];
    x2i = fmaf(xr[c], xr[c], x2i);
  }
  float zs = 0.f;
#pragma unroll
  for (int c = 0; c < D_; ++c) {
    float v = Z[(size_t)i * D_ + c];
    Zs[i * D_ + c] = v;
    zs = fmaf(v, v, zs);
  }
  z2s[i] = zs;
  R[0][i] = BIGF;
  R[1][i] = BIGF;
  __syncthreads();

  float last = BIGF;
  for (int k = 2; k <= 2 * T_; ++k) {
    const int cur = k % 3, p1 = (k - 1) % 3, p2 = (k - 2) % 3;
    const int j = k - 2 - i;
    float nv = BIGF;
    if (j >= 0 && j < T_) {
      float rl = R[p1][i];
      float ru = (i == 0) ? BIGF : R[p1][i - 1];
      float rd = (i == 0) ? ((k == 2) ? 0.f : BIGF) : R[p2][i - 1];
      float dot = 0.f;
#pragma unroll
      for (int c = 0; c < D_; ++c) dot = fmaf(xr[c], Zs[j * D_ + c], dot);
      float dv = fmaf(-2.f, dot, x2i + z2s[j]);
      nv = dv + softmin3(ru, rl, rd);
    }
    R[cur][i] = nv;
    __syncthreads();
    last = nv;
  }
  if (i == T_ - 1) partials[n] = w[n] * last;
}

// ---------------------------------------------------------------------------
// Deterministic final reduction (avoids float-atomic ordering variance).
// ---------------------------------------------------------------------------
__global__ void reduce_kernel(const float* __restrict__ partials,
                              float* __restrict__ out) {
  float s = 0.f;
  for (int n = 0; n < N_; ++n) s += partials[n];
  out[0] = s;
}

extern "C" void kernel_launch(void* const* d_in, const int* in_sizes, int n_in,
                              void* d_out, int out_size, void* d_ws, size_t ws_size,
                              hipStream_t stream) {
  const float* X = (const float*)d_in[0];  // (N, T, d)
  const float* w = (const float*)d_in[1];  // (N,)
  const float* Z = (const float*)d_in[2];  // (T, d)
  float* out = (float*)d_out;
  float* ws  = (float*)d_ws;

  float* partials = ws;  // 64 floats
  const size_t needF = 64 + (size_t)N_ * T_ + T_ + (size_t)N_ * T_ * T_;

  if (ws_size >= needF * sizeof(float)) {
    float* x2  = ws + 64;          // N*T
    float* z2  = x2 + N_ * T_;     // T
    float* Dsk = z2 + T_;          // N*T*T skewed (64 MB, L2-resident)

    const int normN = N_ * T_ + T_;
    norms_kernel<<<(normN + 255) / 256, 256, 0, stream>>>(X, Z, x2, z2);

    dim3 g(T_ / 16, N_);  // 32 row-stripes x 64 batches
    gemm_wmma_kernel<<<g, T_, 0, stream>>>(X, Z, x2, z2, Dsk);

    dp_kernel<<<N_, T_, 0, stream>>>(Dsk, w, partials);
  } else {
    dp_fused_kernel<<<N_, T_, 0, stream>>>(X, Z, w, partials);
  }

  reduce_kernel<<<1, 1, 0, stream>>>(partials, out);
}